// TP_42090679501117
// MI455X (gfx1250) — compile-verified
//
#include <hip/hip_runtime.h>

// MI455X / gfx1250, wave32. Problem is HBM-bound (~328 MB streamed, ~1.3 GFLOP
// after hoisting the h-invariant half of the W_in GEMM). We stream
// neighbor_feats once into LDS via CDNA5 async global->LDS DMA
// (ASYNCcnt-tracked, bypasses VGPRs, overlaps the per-block setup math),
// and use V_WMMA_F32_16X16X4_F32 (full f32) for the per-(n,h,l) projection.

typedef float v2f __attribute__((ext_vector_type(2)));
typedef float v8f __attribute__((ext_vector_type(8)));
typedef int   v4i __attribute__((vector_size(16)));

typedef __attribute__((address_space(1))) v4i* gptr4;  // global int4*
typedef __attribute__((address_space(3))) v4i* lptr4;  // LDS int4*

#define NPTS  10000
#define NH    32
#define NL    4
#define NC    64
#define CMID  8
#define COUT  64
#define ROWPAD 68   // 64 + 4 floats pad -> conflict-free A fragment ds reads

#if defined(__AMDGCN__) && __has_builtin(__builtin_amdgcn_global_load_async_to_lds_b128)
#define HAVE_ASYNC_LDS 1
#else
#define HAVE_ASYNC_LDS 0
#endif

__device__ __forceinline__ void wait_async0() {
#if defined(__AMDGCN__)
#if __has_builtin(__builtin_amdgcn_s_wait_asynccnt)
    __builtin_amdgcn_s_wait_asynccnt(0);
#else
    asm volatile("s_wait_asynccnt 0" ::: "memory");
#endif
#endif
}

__global__ __launch_bounds__(128)
void tp_wigner_kernel(const float* __restrict__ s_points,     // (N+1,3)
                      const float* __restrict__ nbr_points,   // (N,H,3)
                      const float* __restrict__ s_feats,      // (N+1,L,C)
                      const float* __restrict__ nbr_feats,    // (N,H,L,C)
                      const float* __restrict__ W_in,         // (8,128)
                      const float* __restrict__ W_out,        // (64,8)
                      const float* __restrict__ w_m0,         // (2,2)
                      const float* __restrict__ w_m1,         // (2,)
                      float* __restrict__ out)                // (N,4,64)
{
    __shared__ float nf_lds[128 * ROWPAD];  // 128 rows (h*4+l) x 64 cols, padded
    __shared__ float w_lds[64 * 16];        // W_in[:, :64]^T, zero-padded to 16 cols
    __shared__ float sc_lds[64];            // s_contrib[l][m] padded to 16 cols
    __shared__ float red[32];               // h-reduced out[l][m]

    const int n    = blockIdx.x;
    const int t    = threadIdx.x;
    const int lane = t & 31;
    const int wave = t >> 5;
    const int m    = lane & 15;   // WMMA output column (channel), valid if < 8
    const int half = lane >> 4;   // 0: rows 0-7 / K 0-1 ; 1: rows 8-15 / K 2-3

    // ---- stage neighbor_feats[n] (32 KB) into LDS ----
    // Async path: direct global->LDS DMA per lane, ASYNCcnt tracked; the
    // setup math below (independent of nf_lds) overlaps the transfer.
    {
        float* src = const_cast<float*>(nbr_feats) + (size_t)n * (NH * NL * NC);
#pragma unroll
        for (int i = 0; i < 16; ++i) {
            int v4  = i * 128 + t;          // float4 index, 0..2047
            int row = v4 >> 4;              // 16 float4 per 64-float row
            int c4  = (v4 & 15) << 2;
#if HAVE_ASYNC_LDS
            __builtin_amdgcn_global_load_async_to_lds_b128(
                (gptr4)(src + v4 * 4),
                (lptr4)&nf_lds[row * ROWPAD + c4],
                0, 0);
#else
            float4 d4 = ((const float4*)src)[v4];
            *(float4*)&nf_lds[row * ROWPAD + c4] = d4;
#endif
        }
    }
    // ---- stage W_in[:, :64]^T zero-padded (B matrix source) ----
    for (int i = t; i < 64 * 16; i += 128) {
        int k = i >> 4, mm = i & 15;
        w_lds[i] = (mm < 8) ? W_in[mm * 128 + k] : 0.0f;
    }
    if (t < 32) red[t] = 0.0f;
    // ---- h-invariant contribution: s_feats[n,l,:] . W_in[m,64:] ----
    if (t < 64) {
        int l = t >> 4, mm = t & 15;
        float s = 0.0f;
        if (mm < 8) {
            const float* sf = s_feats + (size_t)n * (NL * NC) + l * NC;
            const float* wr = W_in + mm * 128 + 64;
#pragma unroll 8
            for (int c = 0; c < NC; ++c) s = fmaf(sf[c], wr[c], s);
        }
        sc_lds[t] = s;
    }
#if HAVE_ASYNC_LDS
    wait_async0();
#endif
    __syncthreads();

    // ---- preload B fragments: bf[j] covers K = 4j..4j+3 ----
    // B 4x16 layout (mirrors A/C-D striping): VGPR0 lanes0-15 = row K=0,
    // lanes16-31 = row K=2; VGPR1 = K=1 / K=3.
    v2f bf[16];
#pragma unroll
    for (int j = 0; j < 16; ++j) {
        int k0 = 4 * j + 2 * half;
        bf[j].x = w_lds[k0 * 16 + m];
        bf[j].y = w_lds[(k0 + 1) * 16 + m];
    }

    const float wr  = w_m1[0], wi = w_m1[1];
    const float m00 = w_m0[0], m01 = w_m0[1], m10 = w_m0[2], m11 = w_m0[3];
    const float spx = s_points[n * 3 + 0];
    const float spy = s_points[n * 3 + 1];
    const float spz = s_points[n * 3 + 2];

    float acc0 = 0.f, acc1 = 0.f, acc2 = 0.f, acc3 = 0.f;

    // Each wave: 2 tiles of 16 rows (= 4 h's x 4 l's), 8 tiles = 32 h total.
    for (int tsel = 0; tsel < 2; ++tsel) {
        const int tt = wave * 2 + tsel;

        // C init = s_contrib: D row (v + 8*half) has l = v&3, channel = m
        v8f d;
#pragma unroll
        for (int v = 0; v < 8; ++v) d[v] = sc_lds[(v & 3) * 16 + m];

        // A 16x4 f32 layout: lane holds row = lane&15, K pair (2*half + {0,1})
        const float* arow = &nf_lds[(tt * 16 + m) * ROWPAD + 2 * half];
#pragma unroll
        for (int j = 0; j < 16; ++j) {
            v2f a = *(const v2f*)(arow + 4 * j);
            d = __builtin_amdgcn_wmma_f32_16x16x4_f32(
                    false, a, false, bf[j], (short)0, d, false, false);
        }

        // Per-lane Wigner math: this lane owns 2 h's (g=0,1), channel m.
#pragma unroll
        for (int g = 0; g < 2; ++g) {
            const int h = tt * 4 + half * 2 + g;
            const float* np = nbr_points + ((size_t)n * NH + h) * 3;
            float vx = np[0] - spx, vy = np[1] - spy, vz = np[2] - spz;
            float inv = 1.0f / (sqrtf(vx * vx + vy * vy + vz * vz) + 1e-8f);
            vx *= inv; vy *= inv; vz *= inv;
            float cb  = fminf(1.0f, fmaxf(-1.0f, vz));
            float sb  = sqrtf(fmaxf(0.0f, 1.0f - cb * cb));
            float rxy = sqrtf(vx * vx + vy * vy);
            float ir  = (rxy > 1e-20f) ? (1.0f / rxy) : 0.0f;
            float ca  = (rxy > 1e-20f) ? vx * ir : 1.0f;
            float sa  = vy * ir;
            // D1 = R[perm,:][:,perm], perm=[1,2,0]
            const float D00 = ca,      D01 = 0.0f, D02 = -sa;
            const float D10 = sb * sa, D11 = cb,   D12 = sb * ca;
            const float D20 = cb * sa, D21 = -sb,  D22 = cb * ca;

            float q0 = d[4 * g + 0], q1 = d[4 * g + 1];
            float q2 = d[4 * g + 2], q3 = d[4 * g + 3];
            float b0 = D00 * q1 + D01 * q2 + D02 * q3;
            float b1 = D10 * q1 + D11 * q2 + D12 * q3;
            float b2 = D20 * q1 + D21 * q2 + D22 * q3;
            float o0 = m00 * q0 + m01 * b1;
            float o1 = m10 * q0 + m11 * b1;
            float op = wr * b2 - wi * b0;
            float om = wi * b2 + wr * b0;
            // back = D1^T * [om, o1, op]
            float r0 = D00 * om + D10 * o1 + D20 * op;
            float r1 = D01 * om + D11 * o1 + D21 * op;
            float r2 = D02 * om + D12 * o1 + D22 * op;
            acc0 += o0; acc1 += r0; acc2 += r1; acc3 += r2;
        }
    }

    // ---- reduce over h (valid channels only) ----
    if (m < CMID) {
        atomicAdd(&red[0 * 8 + m], acc0);
        atomicAdd(&red[1 * 8 + m], acc1);
        atomicAdd(&red[2 * 8 + m], acc2);
        atomicAdd(&red[3 * 8 + m], acc3);
    }
    __syncthreads();

    // ---- final projection: out[n,l,o] = sum_m red[l][m] * W_out[o][m] ----
    {
        const int o = t & 63;
        const int lb = t >> 6;  // 0 or 1
#pragma unroll
        for (int li = 0; li < 2; ++li) {
            int l = lb * 2 + li;
            float s = 0.0f;
#pragma unroll
            for (int mm = 0; mm < CMID; ++mm)
                s = fmaf(red[l * 8 + mm], W_out[o * 8 + mm], s);
            out[(size_t)n * 256 + l * 64 + o] = s;
        }
    }
}

extern "C" void kernel_launch(void* const* d_in, const int* in_sizes, int n_in,
                              void* d_out, int out_size, void* d_ws, size_t ws_size,
                              hipStream_t stream) {
    (void)in_sizes; (void)n_in; (void)out_size; (void)d_ws; (void)ws_size;
    const float* s_points   = (const float*)d_in[0];
    const float* nbr_points = (const float*)d_in[1];
    const float* s_feats    = (const float*)d_in[2];
    const float* nbr_feats  = (const float*)d_in[3];
    // d_in[4] = values: unused by the reference
    const float* W_in       = (const float*)d_in[5];
    const float* W_out      = (const float*)d_in[6];
    const float* w_m0       = (const float*)d_in[7];
    const float* w_m1       = (const float*)d_in[8];
    float* out = (float*)d_out;

    tp_wigner_kernel<<<NPTS, 128, 0, stream>>>(
        s_points, nbr_points, s_feats, nbr_feats, W_in, W_out, w_m0, w_m1, out);
}